// GraphSAGE_19086834663641
// MI455X (gfx1250) — compile-verified
//
#include <hip/hip_runtime.h>

// GraphSAGE (3x SAGEConv mean-aggr + FC) for MI455X / gfx1250.
//
// Strategy (see analysis): L2-resident gather problem; CSR-by-dst built once
// (int atomics only), wave-per-node coalesced gather-reduce, and exact-f32
// WMMA (v_wmma_f32_16x16x4_f32) for all dense matmuls.

#define N_NODES 100000
#define D_H 64
#define D_OUT 32

typedef __attribute__((ext_vector_type(2))) float v2f;
typedef __attribute__((ext_vector_type(8))) float v8f;

__device__ __forceinline__ v8f wmma_f32_16x16x4(v2f a, v2f b, v8f c) {
  // 8 args: (neg_a, A, neg_b, B, c_mod, C, reuse_a, reuse_b)
  return __builtin_amdgcn_wmma_f32_16x16x4_f32(false, a, false, b, (short)0, c,
                                               false, false);
}

// ---------------------------------------------------------------- CSR build
__global__ void count_deg_kernel(const int* __restrict__ dst, int n_e,
                                 int* __restrict__ counts) {
  int e = blockIdx.x * blockDim.x + threadIdx.x;
  if (e < n_e) atomicAdd(&counts[dst[e]], 1);
}

// Single-block exclusive scan over counts -> offsets[0..n], offsets[0]=0.
__global__ void scan_kernel(const int* __restrict__ counts,
                            int* __restrict__ offsets, int n) {
  __shared__ int tile[1024];
  int carry = 0;
  for (int base = 0; base < n; base += 1024) {
    int i = base + (int)threadIdx.x;
    int v = (i < n) ? counts[i] : 0;
    tile[threadIdx.x] = v;
    __syncthreads();
    for (int off = 1; off < 1024; off <<= 1) {
      int t = (threadIdx.x >= (unsigned)off) ? tile[threadIdx.x - off] : 0;
      __syncthreads();
      tile[threadIdx.x] += t;
      __syncthreads();
    }
    int incl = tile[threadIdx.x];
    if (i < n) offsets[i + 1] = carry + incl;
    carry += tile[1023];
    __syncthreads();
  }
  if (threadIdx.x == 0) offsets[0] = 0;
}

__global__ void deg_inv_kernel(const int* __restrict__ counts,
                               float* __restrict__ deg_inv, int n) {
  int i = blockIdx.x * blockDim.x + threadIdx.x;
  if (i < n) {
    int c = counts[i];
    deg_inv[i] = 1.0f / (float)(c > 0 ? c : 1);
  }
}

__global__ void fill_csr_kernel(const int* __restrict__ src,
                                const int* __restrict__ dst, int n_e,
                                const int* __restrict__ offsets,
                                int* __restrict__ cursor,
                                int* __restrict__ csr) {
  int e = blockIdx.x * blockDim.x + threadIdx.x;
  if (e < n_e) {
    int d = dst[e];
    int pos = atomicAdd(&cursor[d], 1);
    csr[offsets[d] + pos] = src[e];
  }
}

// ------------------------------------------------------- mean aggregation
// One wave per destination node; lane owns float2 of the 64-wide feature row.
// Neighbor row reads are fully coalesced (32 lanes x 8B = 256B contiguous).
__global__ void aggregate_kernel(const float* __restrict__ X,
                                 const int* __restrict__ csr,
                                 const int* __restrict__ offsets,
                                 const float* __restrict__ deg_inv,
                                 float* __restrict__ agg) {
  int lane = threadIdx.x & 31;
  int node = blockIdx.x * (blockDim.x >> 5) + (threadIdx.x >> 5);
  if (node >= N_NODES) return;
  int jb = offsets[node], je = offsets[node + 1];
  float ax = 0.0f, ay = 0.0f;
  const float* Xl = X + 2 * lane;
  int j = jb;
  for (; j + 1 < je; j += 2) {
    int s0 = csr[j];
    int s1 = csr[j + 1];
    if (j + 2 < je)  // gfx1250 global_prefetch_b8 for the row after next
      __builtin_prefetch(Xl + (size_t)csr[j + 2] * D_H, 0, 1);
    v2f v0 = *(const v2f*)(Xl + (size_t)s0 * D_H);
    v2f v1 = *(const v2f*)(Xl + (size_t)s1 * D_H);
    ax += v0.x + v1.x;
    ay += v0.y + v1.y;
  }
  if (j < je) {
    v2f v0 = *(const v2f*)(Xl + (size_t)csr[j] * D_H);
    ax += v0.x;
    ay += v0.y;
  }
  float di = deg_inv[node];
  v2f r;
  r.x = ax * di;
  r.y = ay * di;
  *(v2f*)(agg + (size_t)node * D_H + 2 * lane) = r;
}

// ------------------------------------------------------- SAGE dense stage
// One wave: 16 nodes x 64 outputs.
//   Hout = relu(Agg @ Wl^T + Hin @ Wr^T + bl)
// A layout (16x4 f32): lanes 0-15 -> (M=lane, K=k0+0..1); lanes 16-31 ->
// (M=lane-16, K=k0+2..3). B = W^T chunk uses the mirrored layout:
// B[k][n] = W[n][k], n = lane&15, same K split by lane half.
__global__ void sage_gemm_kernel(const float* __restrict__ agg,
                                 const float* __restrict__ Hin,
                                 const float* __restrict__ Wl,
                                 const float* __restrict__ bl,
                                 const float* __restrict__ Wr,
                                 float* __restrict__ Hout) {
  int lane = threadIdx.x & 31;
  int gw = blockIdx.x * (blockDim.x >> 5) + (threadIdx.x >> 5);
  int node0 = gw * 16;
  if (node0 >= N_NODES) return;  // wave-uniform: EXEC stays all-ones for WMMA

  int l15 = lane & 15;
  int half = lane >> 4;
  const float* aRow = agg + (size_t)(node0 + l15) * D_H + 2 * half;
  const float* hRow = Hin + (size_t)(node0 + l15) * D_H + 2 * half;
  const float* wlRow = Wl + (size_t)l15 * D_H + 2 * half;
  const float* wrRow = Wr + (size_t)l15 * D_H + 2 * half;

  v8f acc[4] = {};
#pragma unroll
  for (int k0 = 0; k0 < D_H; k0 += 4) {
    v2f aA = *(const v2f*)(aRow + k0);
    v2f aH = *(const v2f*)(hRow + k0);
#pragma unroll
    for (int t = 0; t < 4; ++t) {
      v2f bL = *(const v2f*)(wlRow + (size_t)t * 16 * D_H + k0);
      v2f bR = *(const v2f*)(wrRow + (size_t)t * 16 * D_H + k0);
      acc[t] = wmma_f32_16x16x4(aA, bL, acc[t]);
      acc[t] = wmma_f32_16x16x4(aH, bR, acc[t]);
    }
  }

#pragma unroll
  for (int t = 0; t < 4; ++t) {
    float bv = bl[t * 16 + l15];
#pragma unroll
    for (int r = 0; r < 8; ++r) {
      float v = acc[t][r] + bv;
      v = fmaxf(v, 0.0f);  // relu
      int m = r + half * 8;
      Hout[(size_t)(node0 + m) * D_H + t * 16 + l15] = v;
    }
  }
}

// ------------------------------------------------------- final FC (64->32)
__global__ void fc_gemm_kernel(const float* __restrict__ Hin,
                               const float* __restrict__ Wfc,
                               const float* __restrict__ bfc,
                               float* __restrict__ Out) {
  int lane = threadIdx.x & 31;
  int gw = blockIdx.x * (blockDim.x >> 5) + (threadIdx.x >> 5);
  int node0 = gw * 16;
  if (node0 >= N_NODES) return;

  int l15 = lane & 15;
  int half = lane >> 4;
  const float* hRow = Hin + (size_t)(node0 + l15) * D_H + 2 * half;
  const float* wRow = Wfc + (size_t)l15 * D_H + 2 * half;

  v8f acc[2] = {};
#pragma unroll
  for (int k0 = 0; k0 < D_H; k0 += 4) {
    v2f aH = *(const v2f*)(hRow + k0);
#pragma unroll
    for (int t = 0; t < 2; ++t) {
      v2f b = *(const v2f*)(wRow + (size_t)t * 16 * D_H + k0);
      acc[t] = wmma_f32_16x16x4(aH, b, acc[t]);
    }
  }

#pragma unroll
  for (int t = 0; t < 2; ++t) {
    float bv = bfc[t * 16 + l15];
#pragma unroll
    for (int r = 0; r < 8; ++r) {
      float v = acc[t][r] + bv;
      int m = r + half * 8;
      Out[(size_t)(node0 + m) * D_OUT + t * 16 + l15] = v;
    }
  }
}

// ------------------------------------------------------------------ driver
extern "C" void kernel_launch(void* const* d_in, const int* in_sizes, int n_in,
                              void* d_out, int out_size, void* d_ws,
                              size_t ws_size, hipStream_t stream) {
  const float* x = (const float*)d_in[0];
  const int* ei = (const int*)d_in[1];
  const int E = in_sizes[1] / 2;
  const int* src = ei;
  const int* dst = ei + E;
  const float* W1l = (const float*)d_in[2];
  const float* b1l = (const float*)d_in[3];
  const float* W1r = (const float*)d_in[4];
  const float* W2l = (const float*)d_in[5];
  const float* b2l = (const float*)d_in[6];
  const float* W2r = (const float*)d_in[7];
  const float* W3l = (const float*)d_in[8];
  const float* b3l = (const float*)d_in[9];
  const float* W3r = (const float*)d_in[10];
  const float* Wfc = (const float*)d_in[11];
  const float* bfc = (const float*)d_in[12];
  float* out = (float*)d_out;

  // Workspace partition (~85 MB total)
  char* ws = (char*)d_ws;
  size_t off = 0;
  auto take = [&](size_t bytes) -> void* {
    void* p = (void*)(ws + off);
    off += (bytes + 255) & ~(size_t)255;
    return p;
  };
  int* counts = (int*)take((size_t)N_NODES * sizeof(int));
  int* offsets = (int*)take(((size_t)N_NODES + 1) * sizeof(int));
  int* cursor = (int*)take((size_t)N_NODES * sizeof(int));
  int* csr = (int*)take((size_t)E * sizeof(int));
  float* deg_inv = (float*)take((size_t)N_NODES * sizeof(float));
  float* agg = (float*)take((size_t)N_NODES * D_H * sizeof(float));
  float* hA = (float*)take((size_t)N_NODES * D_H * sizeof(float));
  float* hB = (float*)take((size_t)N_NODES * D_H * sizeof(float));

  const int TB = 256;
  const int egrid = (E + TB - 1) / TB;
  const int ngrid = (N_NODES + TB - 1) / TB;
  const int aggGrid = (N_NODES + 7) / 8;            // 8 waves/block, 1 node/wave
  const int gemmGrid = (N_NODES / 16 + 7) / 8;      // 8 waves/block, 16 nodes/wave

  // CSR build (once, shared by all three conv layers)
  hipMemsetAsync(counts, 0, (size_t)N_NODES * sizeof(int), stream);
  hipMemsetAsync(cursor, 0, (size_t)N_NODES * sizeof(int), stream);
  count_deg_kernel<<<egrid, TB, 0, stream>>>(dst, E, counts);
  scan_kernel<<<1, 1024, 0, stream>>>(counts, offsets, N_NODES);
  deg_inv_kernel<<<ngrid, TB, 0, stream>>>(counts, deg_inv, N_NODES);
  fill_csr_kernel<<<egrid, TB, 0, stream>>>(src, dst, E, offsets, cursor, csr);

  // Layer 1: x -> hA
  aggregate_kernel<<<aggGrid, TB, 0, stream>>>(x, csr, offsets, deg_inv, agg);
  sage_gemm_kernel<<<gemmGrid, TB, 0, stream>>>(agg, x, W1l, b1l, W1r, hA);
  // Layer 2: hA -> hB
  aggregate_kernel<<<aggGrid, TB, 0, stream>>>(hA, csr, offsets, deg_inv, agg);
  sage_gemm_kernel<<<gemmGrid, TB, 0, stream>>>(agg, hA, W2l, b2l, W2r, hB);
  // Layer 3: hB -> hA
  aggregate_kernel<<<aggGrid, TB, 0, stream>>>(hB, csr, offsets, deg_inv, agg);
  sage_gemm_kernel<<<gemmGrid, TB, 0, stream>>>(agg, hB, W3l, b3l, W3r, hA);
  // FC: hA -> out
  fc_gemm_kernel<<<gemmGrid, TB, 0, stream>>>(hA, Wfc, bfc, out);
}